// Positional_Encoding_43250320671106
// MI455X (gfx1250) — compile-verified
//
#include <hip/hip_runtime.h>
#include <math.h>

// Problem constants from the reference.
#define SEQ_LEN 4096
#define D_MODEL 1024
#define BATCH   8

typedef float float4v __attribute__((ext_vector_type(4)));

// out[b, s, d] = x[b, s, d] + pe[s, d]
// pe[s, 2i]   = sin(s / 10000^(2i/D))
// pe[s, 2i+1] = cos(s / 10000^(2i/D))
//
// One thread owns one float4 (4 consecutive dims) of one sequence position and
// applies its PE quad to all 8 batch slices (transcendentals amortized 8x).
//
// PE generation uses native CDNA5 trans ops directly:
//   v_sin_f32 / v_cos_f32 compute sin/cos(2*pi*x) with x in revolutions,
//   so we reduce the angle to turns r in [-0.5, 0.5] with one rndne+sub.
// This avoids libm's Payne-Hanek slow path (the ~280-VALU branchy reduction
// seen in the previous round) entirely: ~15 branch-free VALU ops per thread.
__global__ __launch_bounds__(256) void pe_add_kernel(const float* __restrict__ x,
                                                     float* __restrict__ out) {
    const int idx = blockIdx.x * 256 + threadIdx.x;   // 0 .. (4096*1024/4 - 1)
    const int pos = idx >> 8;                         // sequence position (256 float4 per row)
    const int t   = idx & 255;                        // float4 slot within the row
    const int d0  = t << 2;                           // first dim of this float4 (even)

    // inv_freq(even) = 10000^(-even/D) = exp2(-(even/D) * log2(10000))
    const float L2_10K_OVER_D = 13.28771237954945f / (float)D_MODEL;  // log2(10000)/D
    const float INV_2PI       = 0.15915494309189535f;                 // 1/(2*pi)
    const float fpos = (float)pos;

    const float f0 = __builtin_amdgcn_exp2f(-(float)(d0)     * L2_10K_OVER_D); // v_exp_f32
    const float f1 = __builtin_amdgcn_exp2f(-(float)(d0 + 2) * L2_10K_OVER_D);

    // Angle in revolutions, reduced to [-0.5, 0.5].
    float t0 = fpos * f0 * INV_2PI;
    float t1 = fpos * f1 * INV_2PI;
    t0 -= __builtin_rintf(t0);   // v_rndne_f32 + v_sub_f32
    t1 -= __builtin_rintf(t1);

    const float4v pe = {
        __builtin_amdgcn_sinf(t0),   // v_sin_f32: sin(2*pi*t0)
        __builtin_amdgcn_cosf(t0),   // v_cos_f32: cos(2*pi*t0)
        __builtin_amdgcn_sinf(t1),
        __builtin_amdgcn_cosf(t1)
    };

    const size_t off          = (size_t)idx * 4;              // float offset in one batch slice
    const size_t batch_stride = (size_t)SEQ_LEN * (size_t)D_MODEL;

    #pragma unroll
    for (int b = 0; b < BATCH; ++b) {
        const float4v* src = (const float4v*)(x + (size_t)b * batch_stride + off);
        float4v*       dst = (float4v*)(out + (size_t)b * batch_stride + off);
        float4v v = __builtin_nontemporal_load(src);   // global_load_b128 th:NT
        v += pe;
        __builtin_nontemporal_store(v, dst);           // global_store_b128 th:NT
    }
}

extern "C" void kernel_launch(void* const* d_in, const int* in_sizes, int n_in,
                              void* d_out, int out_size, void* d_ws, size_t ws_size,
                              hipStream_t stream) {
    (void)in_sizes; (void)n_in; (void)d_ws; (void)ws_size; (void)out_size;

    const float* x   = (const float*)d_in[0];
    float*       out = (float*)d_out;

    // One thread per float4 of one batch slice: (4096*1024/4) threads.
    const int threads_total = (SEQ_LEN * D_MODEL) / 4;   // 1,048,576
    const int block = 256;                                // 8 wave32s
    const int grid  = threads_total / block;              // 4096 blocks

    pe_add_kernel<<<grid, block, 0, stream>>>(x, out);
}